// PointsRasterizer_86191403696481
// MI455X (gfx1250) — compile-verified
//
#include <hip/hip_runtime.h>
#include <stdint.h>

// ---------------- problem constants (from reference) ----------------
#define HH 512
#define WW 512
#define BB 2
#define CC 8
#define RR 4
#define TILE 16
#define TXN 32              // tiles per row (512/16)
#define NT  1024            // tiles per batch image
#define NTT 2048            // total tiles (B * NT)
#define ZNEARV 0.1f
#define ZFARV  10.0f
#define EPSV   1e-8f
#define NPTS 128            // points staged in LDS per chunk (multiple of 8)
#define NCOL 16             // B-matrix column record size per point

typedef float v2f __attribute__((ext_vector_type(2)));
typedef float v8f __attribute__((ext_vector_type(8)));

__device__ __forceinline__ int iabs_(int v) { return v < 0 ? -v : v; }

// ---- CDNA5 async global->LDS staging (ASYNCcnt-tracked DMA-style copy) ----
typedef __attribute__((address_space(3))) void lds_v;
__device__ __forceinline__ uint32_t lds_off_(void* p) {
    return (uint32_t)(uintptr_t)(lds_v*)p;          // generic -> LDS byte offset
}
__device__ __forceinline__ void async_lds_b128(const void* g, void* lds) {
    asm volatile("global_load_async_to_lds_b128 %0, %1, off"
                 :: "v"(lds_off_(lds)), "v"((uint64_t)(uintptr_t)g)
                 : "memory");
}
__device__ __forceinline__ void wait_asynccnt0() {
    asm volatile("s_wait_asynccnt 0x0" ::: "memory");
}

// ---------------- kernel 1: count fragments per tile ----------------
__global__ void pr_count(const float* __restrict__ pts, int P, int* __restrict__ cnt) {
    int i = blockIdx.x * blockDim.x + threadIdx.x;
    if (i >= BB * P) return;
    int b = i / P;
    float x = pts[(size_t)i * 3 + 0];
    float y = pts[(size_t)i * 3 + 1];
    float z = pts[(size_t)i * 3 + 2];
    if (!(z > ZNEARV && z < ZFARV)) return;
    int cx = (int)rintf((x + 1.f) * 0.5f * (WW - 1));
    int cy = (int)rintf((y + 1.f) * 0.5f * (HH - 1));
    int tx0 = max(cx - RR, 0) >> 4, tx1 = min(cx + RR, WW - 1) >> 4;
    int ty0 = max(cy - RR, 0) >> 4, ty1 = min(cy + RR, HH - 1) >> 4;
    for (int ty = ty0; ty <= ty1; ++ty)
        for (int tx = tx0; tx <= tx1; ++tx)
            atomicAdd(&cnt[b * NT + ty * TXN + tx], 1);
}

// ---------------- kernel 2: exclusive scan over 2048 tiles ----------------
__global__ void pr_scan(const int* __restrict__ cnt, int* __restrict__ start,
                        int* __restrict__ cursor) {
    if (threadIdx.x == 0) {
        int s = 0;
        for (int i = 0; i < NTT; ++i) { start[i] = s; s += cnt[i]; }
        start[NTT] = s;
    }
    __syncthreads();
    for (int i = threadIdx.x; i < NTT; i += blockDim.x) cursor[i] = start[i];
}

// ---------------- kernel 3: scatter point indices into tile lists ----------------
__global__ void pr_scatter(const float* __restrict__ pts, int P,
                           int* __restrict__ cursor, int* __restrict__ list) {
    int i = blockIdx.x * blockDim.x + threadIdx.x;
    if (i >= BB * P) return;
    int b = i / P;
    float x = pts[(size_t)i * 3 + 0];
    float y = pts[(size_t)i * 3 + 1];
    float z = pts[(size_t)i * 3 + 2];
    if (!(z > ZNEARV && z < ZFARV)) return;
    int cx = (int)rintf((x + 1.f) * 0.5f * (WW - 1));
    int cy = (int)rintf((y + 1.f) * 0.5f * (HH - 1));
    int tx0 = max(cx - RR, 0) >> 4, tx1 = min(cx + RR, WW - 1) >> 4;
    int ty0 = max(cy - RR, 0) >> 4, ty1 = min(cy + RR, HH - 1) >> 4;
    int p = i - b * P;
    for (int ty = ty0; ty <= ty1; ++ty)
        for (int tx = tx0; tx <= tx1; ++tx) {
            int slot = atomicAdd(&cursor[b * NT + ty * TXN + tx], 1);
            list[slot] = p;
        }
}

// ---------------- kernel 4: per-tile rasterize (WMMA accumulate) ----------------
// Block = 512 threads = 16 waves; wave w owns pixel row w of a 16x16 tile.
// WMMA: D[16 pix x 16 col] += A[16 pix x 4 pts] * B[4 pts x 16 col]
//   cols 0..7 = features, col 8 = z, col 9 = 1 (denom), 10..15 = 0.
__launch_bounds__(512)
__global__ void pr_raster(const float* __restrict__ pts, const float* __restrict__ feats,
                          const float* __restrict__ sigp, const float* __restrict__ maxr_p,
                          int P, const int* __restrict__ start, const int* __restrict__ list,
                          float* __restrict__ out) {
    __shared__ __align__(16) float sP4[NPTS * 4];       // x, y, z, 1/(2*sigma^2)
    __shared__ __align__(8)  int   sCxy[NPTS * 2];      // cx, cy
    __shared__ __align__(16) float sCol[NPTS * NCOL];   // B-matrix column records
    __shared__ float sAcc[TILE][TILE][10];
    __shared__ float sWs[TILE][TILE];

    int bt = blockIdx.x;
    int b  = bt >> 10;
    int ti = bt & 1023;
    int x0 = (ti & 31) * TILE;
    int y0 = (ti >> 5) * TILE;

    int tid  = threadIdx.x;
    int lane = tid & 31;
    int wave = tid >> 5;                 // pixel row within tile
    int half = lane >> 4;                // K-half for WMMA operands
    int mcol = lane & 15;                // pixel-in-row (A) / output column (B)

    float maxr = maxr_p[0];
    float r2   = maxr * maxr;
    int   s0   = start[bt];
    int   cnt  = start[bt + 1] - s0;

    int   pxl = x0 + mcol;
    int   pyl = y0 + wave;
    float qxn = 2.f * (float)pxl / (float)(WW - 1) - 1.f;
    float qyn = 2.f * (float)pyl / (float)(HH - 1) - 1.f;

    // constant columns: col9 = 1 (denom), col10..15 = 0; also keeps B NaN-free
    for (int idx = tid; idx < NPTS * NCOL; idx += 512)
        sCol[idx] = ((idx & 15) == 9) ? 1.f : 0.f;

    // chunk staging: normal loads for point scalars, async DMA for features
    auto stage = [&](int base, int n, int nPad) {
        __syncthreads();
        for (int j = tid; j < nPad; j += 512) {
            if (j < n) {
                int p = list[s0 + base + j];
                size_t o = (size_t)b * P + p;
                float x = pts[o * 3 + 0];
                float y = pts[o * 3 + 1];
                float z = pts[o * 3 + 2];
                float sg = sigp[o];
                float inv = 1.f / (2.f * sg * sg);
                int cx = (int)rintf((x + 1.f) * 0.5f * (WW - 1));
                int cy = (int)rintf((y + 1.f) * 0.5f * (HH - 1));
                ((float4*)sP4)[j] = make_float4(x, y, z, inv);
                ((int2*)sCxy)[j]  = make_int2(cx, cy);
                sCol[j * NCOL + 8] = z;
                const float* gf = feats + o * 8;
                async_lds_b128(gf,     &sCol[j * NCOL + 0]);
                async_lds_b128(gf + 4, &sCol[j * NCOL + 4]);
            } else {                       // sentinel: z=0 fails zok -> w=0
                ((float4*)sP4)[j] = make_float4(0.f, 0.f, 0.f, 0.f);
                ((int2*)sCxy)[j]  = make_int2(0, 0);
            }
        }
        wait_asynccnt0();
        __syncthreads();
    };

    // branchless per-(pixel,point) validity + z
    auto zcand = [&](int k) -> float {
        float4 p4 = ((const float4*)sP4)[k];
        int2   cc = ((const int2*)sCxy)[k];
        float dx = qxn - p4.x, dy = qyn - p4.y;
        float d2 = dx * dx + dy * dy;
        bool ok = (d2 <= r2) & (iabs_(pxl - cc.x) <= RR) & (iabs_(pyl - cc.y) <= RR)
                & (p4.z > ZNEARV) & (p4.z < ZFARV);
        return ok ? p4.z : ZFARV;
    };

    // ---- sweep 1: per-pixel zmin over valid fragments ----
    float zminl = ZFARV;
    for (int base = 0; base < cnt; base += NPTS) {
        int n = min(NPTS, cnt - base);
        int nPad = (n + 7) & ~7;
        stage(base, n, nPad);
        for (int i = 0; i < nPad; i += 4) {      // 2 pair-evals per lane per iter
            zminl = fminf(zminl, zcand(i + half));
            zminl = fminf(zminl, zcand(i + 2 + half));
        }
    }
    zminl = fminf(zminl, __shfl_xor(zminl, 16, 32));   // merge K halves

    // ---- sweep 2: weights + WMMA accumulation ----
    v8f acc = {0.f, 0.f, 0.f, 0.f, 0.f, 0.f, 0.f, 0.f};
    float wsacc = 0.f;
    const float invg = 2.f;                            // 1/GAMMA, exact

    // returns {A-element (w), B-element (column value)} for (pixel, point k)
    auto pair = [&](int k) -> float2 {
        float4 p4 = ((const float4*)sP4)[k];
        int2   cc = ((const int2*)sCxy)[k];
        float dx = qxn - p4.x, dy = qyn - p4.y;
        float d2 = dx * dx + dy * dy;
        bool ok = (d2 <= r2) & (iabs_(pxl - cc.x) <= RR) & (iabs_(pyl - cc.y) <= RR)
                & (p4.z > ZNEARV) & (p4.z < ZFARV);
        float wsv = __expf(-d2 * p4.w);                    // gaussian
        float wv  = wsv * __expf((zminl - p4.z) * invg);   // depth softmax
        wsacc += ok ? wsv : 0.f;
        return make_float2(ok ? wv : 0.f, sCol[k * NCOL + mcol]);
    };

    for (int base = 0; base < cnt; base += NPTS) {
        int n = min(NPTS, cnt - base);
        int nPad = (n + 7) & ~7;
        stage(base, n, nPad);
        for (int i = 0; i < nPad; i += 8) {      // 8 points -> 2 back-to-back WMMAs
            v2f a0, b0, a1, b1;
            float2 r;
            r = pair(i     + half * 2    ); a0[0] = r.x; b0[0] = r.y;
            r = pair(i     + half * 2 + 1); a0[1] = r.x; b0[1] = r.y;
            r = pair(i + 4 + half * 2    ); a1[0] = r.x; b1[0] = r.y;
            r = pair(i + 4 + half * 2 + 1); a1[1] = r.x; b1[1] = r.y;
            acc = __builtin_amdgcn_wmma_f32_16x16x4_f32(
                false, a0, false, b0, (short)0, acc, false, false);
            acc = __builtin_amdgcn_wmma_f32_16x16x4_f32(
                false, a1, false, b1, (short)0, acc, false, false);
        }
    }

    // ---- transpose WMMA D-layout through LDS, then finalize ----
    float wspix = wsacc + __shfl_xor(wsacc, 16, 32);
    __syncthreads();
    if (lane < 16) sWs[wave][lane] = wspix;
#pragma unroll
    for (int v = 0; v < 8; ++v) {
        int mx = v + (half << 3);           // D row: M=v (lanes 0-15) or v+8 (16-31)
        if (mcol < 10) sAcc[wave][mx][mcol] = acc[v];
    }
    __syncthreads();

    if (tid < 256) {
        int fy = tid >> 4, fx = tid & 15;
        float denom = sAcc[fy][fx][9] + EPSV;
        float inv   = 1.f / denom;
        size_t pix  = ((size_t)b * HH + (y0 + fy)) * WW + (x0 + fx);
        float* oc   = out + pix * CC;
#pragma unroll
        for (int c = 0; c < CC; ++c) oc[c] = sAcc[fy][fx][c] * inv;
        out[(size_t)BB * HH * WW * CC + pix]       = sAcc[fy][fx][8] * inv;      // depth
        out[(size_t)BB * HH * WW * (CC + 1) + pix] = 1.f - __expf(-sWs[fy][fx]); // mask
    }
}

// ---------------- host launcher ----------------
extern "C" void kernel_launch(void* const* d_in, const int* in_sizes, int n_in,
                              void* d_out, int out_size, void* d_ws, size_t ws_size,
                              hipStream_t stream) {
    const float* pts   = (const float*)d_in[0];   // [B,P,3]
    const float* feats = (const float*)d_in[1];   // [B,P,8]
    const float* sig   = (const float*)d_in[2];   // [B,P]
    const float* maxr  = (const float*)d_in[3];   // scalar
    int P = in_sizes[2] / BB;

    int* cnt    = (int*)d_ws;
    int* startA = cnt + NTT;
    int* cursor = startA + NTT + 1;
    int* list   = cursor + NTT;                   // worst case 4 * B * P entries

    (void)hipMemsetAsync(cnt, 0, NTT * sizeof(int), stream);

    int total = BB * P;
    pr_count  <<<(total + 255) / 256, 256, 0, stream>>>(pts, P, cnt);
    pr_scan   <<<1, 256, 0, stream>>>(cnt, startA, cursor);
    pr_scatter<<<(total + 255) / 256, 256, 0, stream>>>(pts, P, cursor, list);
    pr_raster <<<NTT, 512, 0, stream>>>(pts, feats, sig, maxr, P, startA, list,
                                        (float*)d_out);
}